// HyperedgeModel_10316511445099
// MI455X (gfx1250) — compile-verified
//
#include <hip/hip_runtime.h>
#include <math.h>

#define NN 32768
#define GG 2048
#define EE 1146880
#define FF 64

typedef float v2f __attribute__((ext_vector_type(2)));
typedef float v8f __attribute__((ext_vector_type(8)));

// ---------------------------------------------------------------------------
// Kernel 1: node MLP  h[n] = W3^T relu(W2^T relu(W1^T [x||u[batch]])) + b3
// One wave per node; activations staged in LDS (cross-lane K-broadcast).
// ---------------------------------------------------------------------------
__global__ __launch_bounds__(256) void mlp_kernel(
    const float* __restrict__ x, const float* __restrict__ u, const int* __restrict__ batch,
    const float* __restrict__ W1, const float* __restrict__ b1,
    const float* __restrict__ W2, const float* __restrict__ b2,
    const float* __restrict__ W3, const float* __restrict__ b3,
    float* __restrict__ hout)
{
  __shared__ float sIn[8][16];
  __shared__ float sAct[8][64];
  const int tid = threadIdx.x, wave = tid >> 5, lane = tid & 31;
  const int n = blockIdx.x * 8 + wave;

  if (lane < 6)       sIn[wave][lane] = x[n * 6 + lane];
  else if (lane < 10) sIn[wave][lane] = u[batch[n] * 4 + (lane - 6)];
  __syncthreads();

  float a0 = b1[lane], a1 = b1[lane + 32];
  #pragma unroll
  for (int k = 0; k < 10; ++k) {
    float v = sIn[wave][k];
    a0 = fmaf(v, W1[k * 64 + lane], a0);
    a1 = fmaf(v, W1[k * 64 + lane + 32], a1);
  }
  a0 = fmaxf(a0, 0.f); a1 = fmaxf(a1, 0.f);
  sAct[wave][lane] = a0; sAct[wave][lane + 32] = a1;
  __syncthreads();

  float t0 = b2[lane], t1 = b2[lane + 32];
  for (int k = 0; k < 64; ++k) {
    float v = sAct[wave][k];
    t0 = fmaf(v, W2[k * 64 + lane], t0);
    t1 = fmaf(v, W2[k * 64 + lane + 32], t1);
  }
  t0 = fmaxf(t0, 0.f); t1 = fmaxf(t1, 0.f);
  __syncthreads();
  sAct[wave][lane] = t0; sAct[wave][lane + 32] = t1;
  __syncthreads();

  float o0 = b3[lane], o1 = b3[lane + 32];
  for (int k = 0; k < 64; ++k) {
    float v = sAct[wave][k];
    o0 = fmaf(v, W3[k * 64 + lane], o0);
    o1 = fmaf(v, W3[k * 64 + lane + 32], o1);
  }
  hout[n * 64 + lane] = o0; hout[n * 64 + lane + 32] = o1;
}

// ---------------------------------------------------------------------------
// Kernel 2: per-graph segment softmax statistics.
// One block per graph; batch_hyper is sorted, so edges of graph g are a
// contiguous [start,end) found by binary search. Recompute x_hyper on the
// fly (h is L2-resident) for pass1 (max) and pass2 (sum of exp).
// ---------------------------------------------------------------------------
__global__ __launch_bounds__(256) void stats_kernel(
    const float* __restrict__ h, const int* __restrict__ hix, const int* __restrict__ bh,
    float* __restrict__ gmx, float* __restrict__ gden)
{
  __shared__ float red[256];
  __shared__ float smx[64];
  __shared__ int   sse[2];
  const int g = blockIdx.x;
  const int tid = threadIdx.x;

  if (tid < 2) {
    int key = g + tid, lo = 0, hi = EE;
    while (lo < hi) { int mid = (lo + hi) >> 1; if (bh[mid] < key) lo = mid + 1; else hi = mid; }
    sse[tid] = lo;
  }
  __syncthreads();
  const int start = sse[0], end = sse[1];
  const int f = tid & 63, row = tid >> 6;  // 4 edge-rows x 64 features

  float m = -__builtin_inff();
  for (int e = start + row; e < end; e += 4) {
    int i0 = hix[e], i1 = hix[EE + e], i2 = hix[2 * EE + e];
    float v = h[i0 * 64 + f] + h[i1 * 64 + f] + h[i2 * 64 + f];
    m = fmaxf(m, v);
  }
  red[tid] = m;
  __syncthreads();
  if (tid < 64)
    smx[tid] = fmaxf(fmaxf(red[tid], red[tid + 64]), fmaxf(red[tid + 128], red[tid + 192]));
  __syncthreads();

  const float mx = smx[f];
  float s = 0.f;
  for (int e = start + row; e < end; e += 4) {
    int i0 = hix[e], i1 = hix[EE + e], i2 = hix[2 * EE + e];
    float v = h[i0 * 64 + f] + h[i1 * 64 + f] + h[i2 * 64 + f];
    s += expf(v - mx);
  }
  __syncthreads();
  red[tid] = s;
  __syncthreads();
  if (tid < 64) {
    gmx[g * 64 + tid]  = smx[tid];
    gden[g * 64 + tid] = red[tid] + red[tid + 64] + red[tid + 128] + red[tid + 192];
  }
}

// ---------------------------------------------------------------------------
// Kernel 3: fused per-edge pipeline, 16 edges per wave, WMMA f32 16x16x4.
//   xh   = gather-sum (LDS tile, stride 72)
//   xhh  = softmax-coef * relu(xh @ weight)          (GEMM1)
//   o1   = relu([xh||xhh] @ V1 + c1)                 (GEMM2, K=128 as 2x64)
//   o2   = relu(o1 @ V2 + c2)                        (GEMM3)
//   out  = sigmoid(o2 @ V3 + c3)
// A operands from LDS; B operands (weights, 64KB total) stream from L2.
// ---------------------------------------------------------------------------
__global__ __launch_bounds__(128) void edge_kernel(
    const float* __restrict__ h, const int* __restrict__ hix, const int* __restrict__ bh,
    const float* __restrict__ gmx, const float* __restrict__ gden,
    const float* __restrict__ Wt,
    const float* __restrict__ V1, const float* __restrict__ c1,
    const float* __restrict__ V2, const float* __restrict__ c2,
    const float* __restrict__ V3, const float* __restrict__ c3,
    float* __restrict__ out)
{
  __shared__ float sA[4][16 * 72];   // xh tile (later reused for o2)
  __shared__ float sB[4][16 * 72];   // xhh tile
  __shared__ float sC[4][16 * 72];   // o1 tile
  __shared__ float sc1[64], sc2[64], sV3[64];

  const int tid = threadIdx.x;
  if (tid < 64) { sc1[tid] = c1[tid]; sc2[tid] = c2[tid]; sV3[tid] = V3[tid]; }

  const int wave = tid >> 5, lane = tid & 31;
  const int lo = lane & 15, hi = lane >> 4;      // WMMA lane decomposition
  const int e0 = (blockIdx.x * 4 + wave) * 16;   // EE divisible by 64
  float* xh  = sA[wave];
  float* xhh = sB[wave];
  float* o1  = sC[wave];

  // ---- gather & sum R=3 member-node embeddings; each lane does 2 columns
  for (int m = 0; m < 16; ++m) {
    int e = e0 + m;
    int i0 = hix[e], i1 = hix[EE + e], i2 = hix[2 * EE + e];
    float2 v0 = ((const float2*)h)[i0 * 32 + lane];
    float2 v1 = ((const float2*)h)[i1 * 32 + lane];
    float2 v2 = ((const float2*)h)[i2 * 32 + lane];
    xh[m * 72 + 2 * lane]     = v0.x + v1.x + v2.x;
    xh[m * 72 + 2 * lane + 1] = v0.y + v1.y + v2.y;
  }
  __syncthreads();

  // ---- GEMM1: xhh = coef * relu(xh @ Wt)
  for (int n0 = 0; n0 < 64; n0 += 16) {
    v8f acc = {0.f, 0.f, 0.f, 0.f, 0.f, 0.f, 0.f, 0.f};
    #pragma unroll
    for (int k0 = 0; k0 < 64; k0 += 4) {
      int ka = k0 + 2 * hi;
      v2f a, b;
      a.x = xh[lo * 72 + ka];        a.y = xh[lo * 72 + ka + 1];
      b.x = Wt[ka * 64 + n0 + lo];   b.y = Wt[(ka + 1) * 64 + n0 + lo];
      acc = __builtin_amdgcn_wmma_f32_16x16x4_f32(false, a, false, b, (short)0, acc, false, false);
    }
    #pragma unroll
    for (int r = 0; r < 8; ++r) {
      int m = r + 8 * hi, f = n0 + lo, e = e0 + m;
      int g = bh[e];
      float t = fmaxf(acc[r], 0.f);
      float coef = expf(xh[m * 72 + f] - gmx[g * 64 + f]) / gden[g * 64 + f];
      xhh[m * 72 + f] = coef * t;
    }
  }
  __syncthreads();

  // ---- GEMM2: o1 = relu([xh||xhh] @ V1 + c1), K=128 split into two halves
  for (int n0 = 0; n0 < 64; n0 += 16) {
    v8f acc = {0.f, 0.f, 0.f, 0.f, 0.f, 0.f, 0.f, 0.f};
    #pragma unroll
    for (int k0 = 0; k0 < 64; k0 += 4) {
      int ka = k0 + 2 * hi;
      v2f a, b;
      a.x = xh[lo * 72 + ka];        a.y = xh[lo * 72 + ka + 1];
      b.x = V1[ka * 64 + n0 + lo];   b.y = V1[(ka + 1) * 64 + n0 + lo];
      acc = __builtin_amdgcn_wmma_f32_16x16x4_f32(false, a, false, b, (short)0, acc, false, false);
    }
    #pragma unroll
    for (int k0 = 0; k0 < 64; k0 += 4) {
      int ka = k0 + 2 * hi;
      v2f a, b;
      a.x = xhh[lo * 72 + ka];             a.y = xhh[lo * 72 + ka + 1];
      b.x = V1[(64 + ka) * 64 + n0 + lo];  b.y = V1[(64 + ka + 1) * 64 + n0 + lo];
      acc = __builtin_amdgcn_wmma_f32_16x16x4_f32(false, a, false, b, (short)0, acc, false, false);
    }
    #pragma unroll
    for (int r = 0; r < 8; ++r) {
      int m = r + 8 * hi, f = n0 + lo;
      o1[m * 72 + f] = fmaxf(acc[r] + sc1[f], 0.f);
    }
  }
  __syncthreads();

  // ---- GEMM3: o2 = relu(o1 @ V2 + c2)   (o2 overwrites xh tile)
  for (int n0 = 0; n0 < 64; n0 += 16) {
    v8f acc = {0.f, 0.f, 0.f, 0.f, 0.f, 0.f, 0.f, 0.f};
    #pragma unroll
    for (int k0 = 0; k0 < 64; k0 += 4) {
      int ka = k0 + 2 * hi;
      v2f a, b;
      a.x = o1[lo * 72 + ka];        a.y = o1[lo * 72 + ka + 1];
      b.x = V2[ka * 64 + n0 + lo];   b.y = V2[(ka + 1) * 64 + n0 + lo];
      acc = __builtin_amdgcn_wmma_f32_16x16x4_f32(false, a, false, b, (short)0, acc, false, false);
    }
    #pragma unroll
    for (int r = 0; r < 8; ++r) {
      int m = r + 8 * hi, f = n0 + lo;
      xh[m * 72 + f] = fmaxf(acc[r] + sc2[f], 0.f);
    }
  }
  __syncthreads();

  // ---- head: o = sigmoid(o2 @ V3 + c3); also emit batch_hyper as second output
  if (lane < 16) {
    int m = lane, e = e0 + m;
    float s = c3[0];
    #pragma unroll
    for (int f = 0; f < 64; ++f) s = fmaf(xh[m * 72 + f], sV3[f], s);
    out[e] = 1.f / (1.f + expf(-s));
    out[EE + e] = (float)bh[e];
  }
}

// ---------------------------------------------------------------------------
extern "C" void kernel_launch(void* const* d_in, const int* in_sizes, int n_in,
                              void* d_out, int out_size, void* d_ws, size_t ws_size,
                              hipStream_t stream) {
  (void)in_sizes; (void)n_in; (void)out_size; (void)ws_size;
  const float* x     = (const float*)d_in[0];
  const float* u     = (const float*)d_in[1];
  const int*   batch = (const int*)d_in[2];
  const int*   hix   = (const int*)d_in[3];   // hyperedge_index [3,E]
  const int*   bh    = (const int*)d_in[4];   // batch_hyper [E], sorted
  // d_in[5] = r (compile-time constant 3)
  const float* W1 = (const float*)d_in[6];
  const float* b1 = (const float*)d_in[7];
  const float* W2 = (const float*)d_in[8];
  const float* b2 = (const float*)d_in[9];
  const float* W3 = (const float*)d_in[10];
  const float* b3 = (const float*)d_in[11];
  const float* Wt = (const float*)d_in[12];   // weight [64,64]
  const float* V1 = (const float*)d_in[13];
  const float* c1 = (const float*)d_in[14];
  const float* V2 = (const float*)d_in[15];
  const float* c2 = (const float*)d_in[16];
  const float* V3 = (const float*)d_in[17];
  const float* c3 = (const float*)d_in[18];
  float* out = (float*)d_out;

  float* h    = (float*)d_ws;               // N*64 floats (8 MB, L2-resident)
  float* gmx  = h + (size_t)NN * 64;        // G*64
  float* gden = gmx + (size_t)GG * 64;      // G*64

  mlp_kernel  <<<NN / 8, 256, 0, stream>>>(x, u, batch, W1, b1, W2, b2, W3, b3, h);
  stats_kernel<<<GG,     256, 0, stream>>>(h, hix, bh, gmx, gden);
  edge_kernel <<<EE / 64, 128, 0, stream>>>(h, hix, bh, gmx, gden, Wt, V1, c1, V2, c2, V3, c3, out);
}